// AttentionController_56109452755622
// MI455X (gfx1250) — compile-verified
//
#include <hip/hip_runtime.h>
#include <hip/hip_bf16.h>

// Problem constants (reference: N=32, T=2048, D=256, H=8)
#define NB 32
#define TT 2048
#define DD 256
#define HH 8

typedef __attribute__((ext_vector_type(2))) float v2f;
typedef __attribute__((ext_vector_type(8))) float v8f;

#define LDS_STRIDE 260   // 260 % 64 == 4 -> conflict-free ds_load_b64 fragment reads
#define T_PER_BLOCK 128  // 8 waves x 16 rows

// -------------------------------------------------------------------------
// Kernel 1: scores[h,n,t] = sum_e tanh( (MK[n] @ Wm[h]^T)[t,e] ) * o_k[n,e]
// WMMA layout: D[M=e, N=t] = sum_d Wm[h][e][d] * mk[n][t][d]
// grid = (T/128, N, H), block = 256 threads (8 wave32)
// -------------------------------------------------------------------------
__global__ __launch_bounds__(256) void
scores_wmma_kernel(const float* __restrict__ o_k,
                   const float* __restrict__ mk,
                   const float* __restrict__ Wm,
                   float* __restrict__ scores)
{
    extern __shared__ float smem[];               // [128*260] mk tile + [256] o_k
    float* lds_mk = smem;
    float* lds_ok = smem + T_PER_BLOCK * LDS_STRIDE;

    const int tblk = blockIdx.x;                  // 0..15  (t0 = tblk*128)
    const int n    = blockIdx.y;                  // 0..31
    const int h    = blockIdx.z;                  // 0..7
    const int tid  = threadIdx.x;                 // 0..255
    const int wave = tid >> 5;                    // 0..7
    const int lane = tid & 31;
    const int l16  = lane & 15;
    const int lhalf = lane >> 4;                  // 0 or 1

    // ---- stage tile of memory_key[n, t0:t0+128, :] into LDS (padded rows) ----
    const float* mk_base = mk + ((size_t)n * TT + (size_t)tblk * T_PER_BLOCK) * DD;
    for (int i = tid; i < T_PER_BLOCK * (DD / 4); i += 256) {
        const int r = i >> 6;                     // row 0..127
        const int c = (i & 63) << 2;              // col 0,4,...,252
        const float4 v = *(const float4*)(mk_base + ((size_t)r << 8) + c);
        float* dst = lds_mk + r * LDS_STRIDE + c;
        v2f lo, hi;
        lo.x = v.x; lo.y = v.y; hi.x = v.z; hi.y = v.w;
        *(v2f*)(dst)     = lo;
        *(v2f*)(dst + 2) = hi;
    }
    lds_ok[tid] = o_k[n * DD + tid];              // tid < 256 == DD
    __syncthreads();

    // ---- per-wave 16x16x4 fp32 WMMA over full K=256, 16 e-tiles resident ----
    // A frag (Wm): lane -> row (e0 + l16), cols k*4 + 2*lhalf + {0,1}
    // B frag (mk): lane -> row (t0 + l16), cols k*4 + 2*lhalf + {0,1}
    const float* aptr   = Wm + (size_t)h * DD * DD + (size_t)l16 * DD + (lhalf << 1);
    const float* ldsrow = lds_mk + (wave * 16 + l16) * LDS_STRIDE + (lhalf << 1);

    v8f acc[16];
    #pragma unroll
    for (int et = 0; et < 16; ++et) acc[et] = (v8f){};

    #pragma unroll 2
    for (int k = 0; k < DD / 4; ++k) {
        const v2f b = *(const v2f*)(ldsrow + (k << 2));
        #pragma unroll
        for (int et = 0; et < 16; ++et) {
            const v2f a = *(const v2f*)(aptr + et * 16 * DD + (k << 2));
            acc[et] = __builtin_amdgcn_wmma_f32_16x16x4_f32(
                false, a, false, b, (short)0, acc[et], false, false);
        }
    }

    // ---- tanh + o_k weighting; lane holds e = et*16 + 8*lhalf + g, t = l16 ----
    float s = 0.0f;
    #pragma unroll
    for (int et = 0; et < 16; ++et) {
        const float* okp = lds_ok + et * 16 + (lhalf << 3);
        #pragma unroll
        for (int g = 0; g < 8; ++g)
            s += tanhf(acc[et][g]) * okp[g];
    }
    s += __shfl_xor(s, 16, 32);                   // combine the two e-halves

    if (lhalf == 0) {
        const size_t row = ((size_t)h * NB + n) * TT;
        scores[row + tblk * T_PER_BLOCK + wave * 16 + l16] = s;
    }
}

// -------------------------------------------------------------------------
// Kernel 2: softmax over T and rep[n,h,d] = sum_t p[t] * MV[n,t,d]
// grid = (N, H), block = 256
// -------------------------------------------------------------------------
__global__ __launch_bounds__(256) void
softmax_rep_kernel(const float* __restrict__ scores,
                   const float* __restrict__ mv,
                   float* __restrict__ rep)
{
    __shared__ float p[TT];
    __shared__ float red[8];
    const int n = blockIdx.x, h = blockIdx.y;
    const int tid = threadIdx.x;
    const float* srow = scores + ((size_t)h * NB + n) * TT;

    float mx = -3.402823466e38f;
    for (int t = tid; t < TT; t += 256) { const float v = srow[t]; p[t] = v; mx = fmaxf(mx, v); }
    #pragma unroll
    for (int o = 16; o > 0; o >>= 1) mx = fmaxf(mx, __shfl_xor(mx, o, 32));
    if ((tid & 31) == 0) red[tid >> 5] = mx;
    __syncthreads();
    float bmx = red[0];
    #pragma unroll
    for (int i = 1; i < 8; ++i) bmx = fmaxf(bmx, red[i]);
    __syncthreads();

    float sum = 0.0f;
    for (int t = tid; t < TT; t += 256) { const float e = __expf(p[t] - bmx); p[t] = e; sum += e; }
    #pragma unroll
    for (int o = 16; o > 0; o >>= 1) sum += __shfl_xor(sum, o, 32);
    if ((tid & 31) == 0) red[tid >> 5] = sum;
    __syncthreads();                              // also makes all p[] writes visible
    float bsum = 0.0f;
    #pragma unroll
    for (int i = 0; i < 8; ++i) bsum += red[i];
    const float inv = 1.0f / bsum;

    // thread tid == feature d; coalesced streaming of MV[n,:,:]
    // 4 independent accumulators to break the serial FMA dependence chain
    float a0 = 0.0f, a1 = 0.0f, a2 = 0.0f, a3 = 0.0f;
    const float* mvb = mv + (size_t)n * TT * DD + tid;
    for (int t = 0; t < TT; t += 4) {
        a0 = fmaf(p[t + 0], mvb[(size_t)(t + 0) * DD], a0);
        a1 = fmaf(p[t + 1], mvb[(size_t)(t + 1) * DD], a1);
        a2 = fmaf(p[t + 2], mvb[(size_t)(t + 2) * DD], a2);
        a3 = fmaf(p[t + 3], mvb[(size_t)(t + 3) * DD], a3);
    }
    const float acc = (a0 + a1) + (a2 + a3);
    rep[((size_t)n * HH + h) * DD + tid] = acc * inv;   // concat layout [N, H*D]
}

// -------------------------------------------------------------------------
// Kernel 3: out[n,e] = sum_j concat[n,j] * Wo_w[e,j] + Wo_b[e]
// grid = N, block = 256 (8 waves, each wave sweeps 32 output rows)
// -------------------------------------------------------------------------
__global__ __launch_bounds__(256) void
outproj_kernel(const float* __restrict__ rep,
               const float* __restrict__ Wo_w,
               const float* __restrict__ Wo_b,
               float* __restrict__ out)
{
    __shared__ float cs[HH * DD];
    const int n = blockIdx.x;
    const int tid = threadIdx.x;
    const int wave = tid >> 5, lane = tid & 31;

    for (int j = tid; j < HH * DD; j += 256) cs[j] = rep[(size_t)n * HH * DD + j];
    __syncthreads();

    for (int e = wave; e < DD; e += 8) {
        const float* w = Wo_w + (size_t)e * HH * DD;
        float acc = 0.0f;
        #pragma unroll 4
        for (int j = lane; j < HH * DD; j += 32)
            acc = fmaf(cs[j], w[j], acc);
        #pragma unroll
        for (int o = 16; o > 0; o >>= 1) acc += __shfl_xor(acc, o, 32);
        if (lane == 0) out[n * DD + e] = acc + Wo_b[e];
    }
}

// -------------------------------------------------------------------------
extern "C" void kernel_launch(void* const* d_in, const int* in_sizes, int n_in,
                              void* d_out, int out_size, void* d_ws, size_t ws_size,
                              hipStream_t stream)
{
    const float* o_k  = (const float*)d_in[0];   // [N, D]
    const float* mk   = (const float*)d_in[1];   // [N, T, D]
    const float* mv   = (const float*)d_in[2];   // [N, T, D]
    const float* Wm   = (const float*)d_in[3];   // [H, D, D]
    const float* Wo_w = (const float*)d_in[4];   // [D, H*D]
    const float* Wo_b = (const float*)d_in[5];   // [D]
    float* out = (float*)d_out;                  // [N, D]

    float* ws     = (float*)d_ws;
    float* scores = ws;                          // H*N*T   = 524288 floats (2 MB)
    float* rep    = ws + (size_t)HH * NB * TT;   // N*H*D   = 65536 floats (256 KB)

    const size_t lds1 = (size_t)(T_PER_BLOCK * LDS_STRIDE + DD) * sizeof(float);
    scores_wmma_kernel<<<dim3(TT / T_PER_BLOCK, NB, HH), 256, lds1, stream>>>(
        o_k, mk, Wm, scores);
    softmax_rep_kernel<<<dim3(NB, HH), 256, 0, stream>>>(scores, mv, rep);
    outproj_kernel<<<dim3(NB), 256, 0, stream>>>(rep, Wo_w, Wo_b, out);
}